// Processor_59768764891683
// MI455X (gfx1250) — compile-verified
//
#include <hip/hip_runtime.h>
#include <math.h>

#define N_NODES 50000
#define N_EDGES 640000
#define HIDDEN  128

typedef __attribute__((ext_vector_type(2))) float v2f;
typedef __attribute__((ext_vector_type(4))) float v4f;
typedef __attribute__((ext_vector_type(8))) float v8f;
typedef __attribute__((ext_vector_type(4))) unsigned v4u;

// Order-preserving float -> uint key for atomic max (handles negatives, -inf)
__device__ __forceinline__ unsigned fkey(float f) {
  unsigned b = __float_as_uint(f);
  return (b & 0x80000000u) ? ~b : (b | 0x80000000u);
}
__device__ __forceinline__ float fdec(unsigned u) {
  unsigned b = (u & 0x80000000u) ? (u & 0x7FFFFFFFu) : ~u;
  return __uint_as_float(b);
}
#define NEG_INF_KEY 0x007FFFFFu  // fkey(-INFINITY)

__global__ __launch_bounds__(256) void init_agg_kernel(unsigned* __restrict__ aggk, int n4) {
  int i = blockIdx.x * blockDim.x + threadIdx.x;
  if (i < n4) {
    v4u v = {NEG_INF_KEY, NEG_INF_KEY, NEG_INF_KEY, NEG_INF_KEY};
    *(v4u*)(aggk + 4 * (long)i) = v;
  }
}

// Repack a row-major [256+,128] weight matrix into WMMA-lane order:
// packed float4 at index (t*2 + half)*128 + n holds
//   { W[8t+2h+0][n], W[8t+2h+1][n], W[8t+2h+4][n], W[8t+2h+5][n] }
// so one b128 load per lane covers B fragments for two consecutive k-steps.
__global__ __launch_bounds__(256) void pack_w_kernel(const float* __restrict__ Wsrc,
                                                     float* __restrict__ Wpk) {
  int idx = blockIdx.x * blockDim.x + threadIdx.x;  // 32*2*128 = 8192
  if (idx >= 32 * 2 * 128) return;
  int n    = idx & 127;
  int half = (idx >> 7) & 1;
  int t    = idx >> 8;
  int Ka   = 8 * t + 2 * half;
  v4f v;
  v.x = Wsrc[(long)(Ka + 0) * HIDDEN + n];
  v.y = Wsrc[(long)(Ka + 1) * HIDDEN + n];
  v.z = Wsrc[(long)(Ka + 4) * HIDDEN + n];
  v.w = Wsrc[(long)(Ka + 5) * HIDDEN + n];
  *(v4f*)(Wpk + 4 * (long)idx) = v;
}

// One wave computes messages for 16 edges:
//   msg[16,128] = [z[src] | z[dst]] @ W_msg[0:256,:]   (WMMA f32 16x16x4, K=256)
//               + w * W_msg[256,:] + b_msg             (VALU epilogue)
// then scatter-max into aggk via native atomic u32-max on order-preserving keys.
__global__ __launch_bounds__(256) void edge_msg_kernel(
    const float* __restrict__ z, const int* __restrict__ src,
    const int* __restrict__ dst, const float* __restrict__ w,
    const float* __restrict__ Wpk, const float* __restrict__ Wm,
    const float* __restrict__ bm, unsigned* __restrict__ aggk)
{
  const int tile = (int)((blockIdx.x * (unsigned)blockDim.x + threadIdx.x) >> 5);
  if (tile >= N_EDGES / 16) return;              // wave-uniform: EXEC stays all-1s
  const int lane = threadIdx.x & 31;
  const int m    = lane & 15;                    // A-matrix row (edge in tile)
  const int half = lane >> 4;                    // K-pair select within 16x4 step
  const int e0   = tile * 16;

  // ---- preload A fragments: 64 k-steps x v2f = 128 VGPRs ----
  const int e = e0 + m;
  const float* pa = z + (long)src[e] * HIDDEN + half * 2;
  const float* pc = z + (long)dst[e] * HIDDEN + half * 2;
  v2f A[64];
#pragma unroll
  for (int k = 0; k < 32; ++k) A[k]      = *(const v2f*)(pa + 4 * k);
#pragma unroll
  for (int k = 0; k < 32; ++k) A[32 + k] = *(const v2f*)(pc + 4 * k);

  // ---- per-lane epilogue metadata: C/D row r = edge e0 + r + half*8 ----
  int   dv[8];
  float wv[8];
#pragma unroll
  for (int r = 0; r < 8; ++r) {
    int er = e0 + r + half * 8;
    dv[r] = dst[er];
    wv[r] = w[er];
  }

  const float* wl = Wm + 256 * HIDDEN;           // weight-column row of W_msg

#pragma unroll 1
  for (int np = 0; np < 4; ++np) {               // 8 N-tiles, 2 at a time for ILP
    const int n0 = np * 2, n1 = np * 2 + 1;
    v8f acc0 = {}, acc1 = {};
    const float* p0 = Wpk + (long)(half * 128 + n0 * 16 + m) * 4;
    const float* p1 = Wpk + (long)(half * 128 + n1 * 16 + m) * 4;
#pragma unroll
    for (int t = 0; t < 32; ++t) {               // 2 k-steps per iteration
      v4f q0 = *(const v4f*)(p0 + t * 1024);     // b128: frags for k=2t, 2t+1 (n0)
      v4f q1 = *(const v4f*)(p1 + t * 1024);     // b128: frags for k=2t, 2t+1 (n1)
      v2f b00; b00.x = q0.x; b00.y = q0.y;
      v2f b01; b01.x = q0.z; b01.y = q0.w;
      v2f b10; b10.x = q1.x; b10.y = q1.y;
      v2f b11; b11.x = q1.z; b11.y = q1.w;
      acc0 = __builtin_amdgcn_wmma_f32_16x16x4_f32(false, A[2 * t],     false, b00,
                                                   (short)0, acc0, false, false);
      acc1 = __builtin_amdgcn_wmma_f32_16x16x4_f32(false, A[2 * t],     false, b10,
                                                   (short)0, acc1, false, false);
      acc0 = __builtin_amdgcn_wmma_f32_16x16x4_f32(false, A[2 * t + 1], false, b01,
                                                   (short)0, acc0, false, false);
      acc1 = __builtin_amdgcn_wmma_f32_16x16x4_f32(false, A[2 * t + 1], false, b11,
                                                   (short)0, acc1, false, false);
    }
#pragma unroll
    for (int tt = 0; tt < 2; ++tt) {
      v8f acc  = tt ? acc1 : acc0;
      int col  = (tt ? n1 : n0) * 16 + m;
      float bias = bm[col];
      float wrow = wl[col];
#pragma unroll
      for (int r = 0; r < 8; ++r) {              // C/D row r -> edge e0 + r + half*8
        float val = acc[r] + bias + wv[r] * wrow;
        atomicMax(&aggk[(long)dv[r] * HIDDEN + col], fkey(val));
      }
    }
  }
}

// One wave per tile of 16 nodes: out[16,128] = [z | agg] @ W_upd + b_upd
__global__ __launch_bounds__(256) void node_upd_kernel(
    const float* __restrict__ z, const unsigned* __restrict__ aggk,
    const float* __restrict__ Wpk, const float* __restrict__ bu,
    float* __restrict__ out)
{
  const int tile = (int)((blockIdx.x * (unsigned)blockDim.x + threadIdx.x) >> 5);
  if (tile >= N_NODES / 16) return;              // wave-uniform
  const int lane = threadIdx.x & 31;
  const int m    = lane & 15;
  const int half = lane >> 4;
  const int v0   = tile * 16;

  const int node = v0 + m;
  const float*    pz = z    + (long)node * HIDDEN + half * 2;
  const unsigned* pg = aggk + (long)node * HIDDEN + half * 2;
  v2f A[64];
#pragma unroll
  for (int k = 0; k < 32; ++k) A[k] = *(const v2f*)(pz + 4 * k);
#pragma unroll
  for (int k = 0; k < 32; ++k) {                 // decode agg keys, isinf -> 0
    unsigned u0 = pg[4 * k], u1 = pg[4 * k + 1];
    float f0 = fdec(u0), f1 = fdec(u1);
    if (isinf(f0)) f0 = 0.0f;
    if (isinf(f1)) f1 = 0.0f;
    v2f a; a.x = f0; a.y = f1;
    A[32 + k] = a;
  }

#pragma unroll 1
  for (int np = 0; np < 4; ++np) {
    const int n0 = np * 2, n1 = np * 2 + 1;
    v8f acc0 = {}, acc1 = {};
    const float* p0 = Wpk + (long)(half * 128 + n0 * 16 + m) * 4;
    const float* p1 = Wpk + (long)(half * 128 + n1 * 16 + m) * 4;
#pragma unroll
    for (int t = 0; t < 32; ++t) {
      v4f q0 = *(const v4f*)(p0 + t * 1024);
      v4f q1 = *(const v4f*)(p1 + t * 1024);
      v2f b00; b00.x = q0.x; b00.y = q0.y;
      v2f b01; b01.x = q0.z; b01.y = q0.w;
      v2f b10; b10.x = q1.x; b10.y = q1.y;
      v2f b11; b11.x = q1.z; b11.y = q1.w;
      acc0 = __builtin_amdgcn_wmma_f32_16x16x4_f32(false, A[2 * t],     false, b00,
                                                   (short)0, acc0, false, false);
      acc1 = __builtin_amdgcn_wmma_f32_16x16x4_f32(false, A[2 * t],     false, b10,
                                                   (short)0, acc1, false, false);
      acc0 = __builtin_amdgcn_wmma_f32_16x16x4_f32(false, A[2 * t + 1], false, b01,
                                                   (short)0, acc0, false, false);
      acc1 = __builtin_amdgcn_wmma_f32_16x16x4_f32(false, A[2 * t + 1], false, b11,
                                                   (short)0, acc1, false, false);
    }
#pragma unroll
    for (int tt = 0; tt < 2; ++tt) {
      v8f acc = tt ? acc1 : acc0;
      int col = (tt ? n1 : n0) * 16 + m;
      float bias = bu[col];
#pragma unroll
      for (int r = 0; r < 8; ++r) {
        int row = v0 + r + half * 8;
        out[(long)row * HIDDEN + col] = acc[r] + bias;
      }
    }
  }
}

extern "C" void kernel_launch(void* const* d_in, const int* in_sizes, int n_in,
                              void* d_out, int out_size, void* d_ws, size_t ws_size,
                              hipStream_t stream) {
  const float* z   = (const float*)d_in[0];
  const int*   src = (const int*)  d_in[1];
  const int*   dst = (const int*)  d_in[2];
  const float* w   = (const float*)d_in[3];
  const float* Wm  = (const float*)d_in[4];
  const float* bm  = (const float*)d_in[5];
  const float* Wu  = (const float*)d_in[6];
  const float* bu  = (const float*)d_in[7];
  float* out = (float*)d_out;

  // workspace layout: agg keys (25.6 MB) | packed W_msg (128 KB) | packed W_upd (128 KB)
  unsigned* aggk  = (unsigned*)d_ws;
  float*    packM = (float*)((char*)d_ws + (size_t)N_NODES * HIDDEN * 4);
  float*    packU = packM + 32 * 2 * 128 * 4;

  const int aggN4 = (N_NODES * HIDDEN) / 4;      // 1.6M b128 stores
  init_agg_kernel<<<(aggN4 + 255) / 256, 256, 0, stream>>>(aggk, aggN4);

  pack_w_kernel<<<32, 256, 0, stream>>>(Wm, packM);
  pack_w_kernel<<<32, 256, 0, stream>>>(Wu, packU);

  const int wpb    = 256 / 32;                   // 8 waves per block
  const int etiles = N_EDGES / 16;               // 40000
  edge_msg_kernel<<<(etiles + wpb - 1) / wpb, 256, 0, stream>>>(z, src, dst, w,
                                                                packM, Wm, bm, aggk);

  const int ntiles = N_NODES / 16;               // 3125
  node_upd_kernel<<<(ntiles + wpb - 1) / wpb, 256, 0, stream>>>(z, aggk, packU, bu, out);
}